// RenderLayer_10685878632929
// MI455X (gfx1250) — compile-verified
//
#include <hip/hip_runtime.h>

// ---------------------------------------------------------------------------
// Problem constants (match reference)
// ---------------------------------------------------------------------------
#define HH     256
#define WW     256
#define PP     (HH * WW)        // 65536 pixels
#define TEXR_  2048
#define BB_    2
#define VV_    4096
#define FF_    2048
#define TT_    4096
#define EPSA   1e-8f

// d_out flat offsets (float elements), reference tuple order:
// render(B,3,H,W), mask(B,1,H,W), depth(B,H,W), ibuf(B,H,W), vt_img(B,H,W,2), bary(B,3,H,W)
#define OFF_RENDER 0
#define OFF_MASK   (BB_ * 3 * PP)                       // 393216
#define OFF_DEPTH  (OFF_MASK + BB_ * PP)                // 524288
#define OFF_IBUF   (OFF_DEPTH + BB_ * PP)               // 655360
#define OFF_VT     (OFF_IBUF + BB_ * PP)                // 786432
#define OFF_BARY   (OFF_VT + BB_ * PP * 2)              // 1048576

#ifndef __has_builtin
#define __has_builtin(x) 0
#endif

#if __has_builtin(__builtin_amdgcn_global_load_async_to_lds_b32) && \
    __has_builtin(__builtin_amdgcn_s_wait_asynccnt)
#define USE_ASYNC_LDS 1
#else
#define USE_ASYNC_LDS 0
#endif

typedef float v2f __attribute__((ext_vector_type(2)));
typedef float v8f __attribute__((ext_vector_type(8)));

// ---------------------------------------------------------------------------
// Kernel 1: vertex projection via V_WMMA_F32_16X16X4_F32.
// One wave (32 lanes) transforms 16 vertices:  D = A(16x4) * B(4x16)
//   A[m][k] = (v[m] - campos)[k]  (k<3, K=3 padded 0)
//   B[k][n] = camrot[n][k]        (n<3, rest 0)   => D[m][n] = v_cam[m][n]
// A layout: lanes0-15 hold K=0,1 ; lanes16-31 hold K=2,3 (2 VGPRs / lane).
// D layout: VGPR r <-> row M=r+8*half, col N=lane&15. Bounce through LDS to
// get per-vertex values back into lanes, then do the pinhole projection.
// ---------------------------------------------------------------------------
__global__ __launch_bounds__(32)
void proj_wmma_kernel(const float* __restrict__ v,
                      const float* __restrict__ campos,
                      const float* __restrict__ camrot,
                      const float* __restrict__ focal,
                      const float* __restrict__ princpt,
                      float* __restrict__ vpix)
{
    __shared__ float sh[16 * 16];

    const int lane   = threadIdx.x;          // 0..31
    const int gpb    = VV_ / 16;             // vertex groups per batch
    const int b      = blockIdx.x / gpb;
    const int base   = (blockIdx.x % gpb) * 16;
    const int m      = lane & 15;
    const int hf     = lane >> 4;            // 0: K={0,1}, 1: K={2,3}
    const int vid    = base + m;

    // A operand: vertex deltas
    v2f a;
    const size_t vo = ((size_t)b * VV_ + vid) * 3;
    if (hf == 0) {
        a.x = v[vo + 0] - campos[b * 3 + 0];
        a.y = v[vo + 1] - campos[b * 3 + 1];
    } else {
        a.x = v[vo + 2] - campos[b * 3 + 2];
        a.y = 0.0f;
    }

    // B operand: camrot transposed into top-left 3x3, zero elsewhere
    const int n  = m;           // column
    const int k0 = hf * 2;      // K of first VGPR for this half-wave
    v2f bm;
    bm.x = (n < 3 && (k0 + 0) < 3) ? camrot[b * 9 + n * 3 + (k0 + 0)] : 0.0f;
    bm.y = (n < 3 && (k0 + 1) < 3) ? camrot[b * 9 + n * 3 + (k0 + 1)] : 0.0f;

    v8f c = {};
    v8f d = __builtin_amdgcn_wmma_f32_16x16x4_f32(
        /*neg_a=*/false, a, /*neg_b=*/false, bm,
        /*c_mod=*/(short)0, c, /*reuse_a=*/false, /*reuse_b=*/false);

#pragma unroll
    for (int r = 0; r < 8; ++r)
        sh[(r + 8 * hf) * 16 + m] = d[r];     // sh[M*16 + N] = v_cam[M][N]

    __syncthreads();

    if (lane < 16) {
        const float xc = sh[m * 16 + 0];
        const float yc = sh[m * 16 + 1];
        const float zc = sh[m * 16 + 2];
        const float xn = xc / zc;
        const float yn = yc / zc;
        const float pxv = focal[b * 4 + 0] * xn + focal[b * 4 + 1] * yn + princpt[b * 2 + 0];
        const float pyv = focal[b * 4 + 2] * xn + focal[b * 4 + 3] * yn + princpt[b * 2 + 1];
        const size_t o = ((size_t)b * VV_ + vid) * 3;
        vpix[o + 0] = pxv;
        vpix[o + 1] = pyv;
        vpix[o + 2] = zc;
    }
}

// ---------------------------------------------------------------------------
// Kernel 2: fused rasterize + shade.
// One block = 256 pixels of one batch image. All F=2048 triangles' screen
// coords (3 verts x (x,y,z), padded to float4) gathered into 96 KB of LDS
// via the gfx1250 async-to-LDS DMA path, then each thread z-loops all
// triangles (LDS broadcast reads) and shades its winning triangle.
// ---------------------------------------------------------------------------
__global__ __launch_bounds__(256)
void raster_shade_kernel(const float* __restrict__ vpix,
                         const int*   __restrict__ vi,
                         const int*   __restrict__ vti,
                         const float* __restrict__ vt,
                         const float* __restrict__ tex,
                         float*       __restrict__ out)
{
    extern __shared__ float shtri[];          // FF_ * 12 floats (96 KB)

    const int tid = threadIdx.x;
    const int b   = blockIdx.x / (PP / 256);
    const int pix = (blockIdx.x % (PP / 256)) * 256 + tid;
    const float px = (float)(pix % WW);
    const float py = (float)(pix / WW);

    // ---- stage triangle screen data: indexed gather -> LDS (async DMA) ----
    for (int e = tid; e < FF_ * 9; e += 256) {
        const int f  = e / 9;
        const int r  = e - f * 9;
        const int k  = r / 3;
        const int cc = r - k * 3;
        const int vid = vi[f * 3 + k];
        const float* gp = vpix + ((size_t)b * VV_ + vid) * 3 + cc;
        float* lp = shtri + (f * 12 + k * 4 + cc);
#if USE_ASYNC_LDS
        __builtin_amdgcn_global_load_async_to_lds_b32(
            (__attribute__((address_space(1))) int*)gp,
            (__attribute__((address_space(3))) int*)lp, 0, 0);
#else
        *lp = *gp;
#endif
    }
#if USE_ASYNC_LDS
    __builtin_amdgcn_s_wait_asynccnt(0);
#endif
    __syncthreads();

    const float4* sh4 = (const float4*)shtri; // [f*3 + k] = (x,y,z,pad)

    // ---------------- z-buffer loop over all triangles ----------------
    float zmin = __builtin_inff();
    int   best = -1;
    for (int f = 0; f < FF_; ++f) {
        const float4 va = sh4[f * 3 + 0];
        const float4 vb = sh4[f * 3 + 1];
        const float4 vc = sh4[f * 3 + 2];
        const float area = (vb.x - va.x) * (vc.y - va.y) - (vb.y - va.y) * (vc.x - va.x);
        const float w0 = (vc.x - vb.x) * (py - vb.y) - (vc.y - vb.y) * (px - vb.x);
        const float w1 = (va.x - vc.x) * (py - vc.y) - (va.y - vc.y) * (px - vc.x);
        const float w2 = (vb.x - va.x) * (py - va.y) - (vb.y - va.y) * (px - va.x);
        const bool okarea = fabsf(area) > EPSA;
        const float inv = 1.0f / (okarea ? area : 1.0f);
        const float b0 = w0 * inv, b1 = w1 * inv, b2 = w2 * inv;
        const float invz = b0 / va.z + b1 / vb.z + b2 / vc.z;
        const bool valid = (b0 >= 0.0f) & (b1 >= 0.0f) & (b2 >= 0.0f) & okarea &
                           (va.z > 0.0f) & (vb.z > 0.0f) & (vc.z > 0.0f) & (invz > 0.0f);
        const float z = valid ? (1.0f / invz) : __builtin_inff();
        if (z < zmin) { zmin = z; best = f; }
    }

    // ------------------------------ shade ------------------------------
    const int tri = best < 0 ? 0 : best;
    const float4 va = sh4[tri * 3 + 0];
    const float4 vb = sh4[tri * 3 + 1];
    const float4 vc = sh4[tri * 3 + 2];
    const float area = (vb.x - va.x) * (vc.y - va.y) - (vb.y - va.y) * (vc.x - va.x);
    const float w0 = (vc.x - vb.x) * (py - vb.y) - (vc.y - vb.y) * (px - vb.x);
    const float w1 = (va.x - vc.x) * (py - vc.y) - (va.y - vc.y) * (px - vc.x);
    const float w2 = (vb.x - va.x) * (py - va.y) - (vb.y - va.y) * (px - va.x);
    const float inv = 1.0f / (fabsf(area) > EPSA ? area : 1.0f);
    const float b0 = w0 * inv, b1 = w1 * inv, b2 = w2 * inv;
    float invz = b0 / va.z + b1 / vb.z + b2 / vc.z;
    invz = (invz != 0.0f) ? invz : 1.0f;
    const float c0 = b0 / va.z / invz;
    const float c1 = b1 / vb.z / invz;
    const float c2 = b2 / vc.z / invz;

    const int t0 = vti[tri * 3 + 0];
    const int t1 = vti[tri * 3 + 1];
    const int t2 = vti[tri * 3 + 2];
    const float u0 = vt[t0 * 2 + 0], q0 = 1.0f - vt[t0 * 2 + 1];
    const float u1 = vt[t1 * 2 + 0], q1 = 1.0f - vt[t1 * 2 + 1];
    const float u2 = vt[t2 * 2 + 0], q2 = 1.0f - vt[t2 * 2 + 1];
    const float uu = c0 * u0 + c1 * u1 + c2 * u2;
    const float qq = c0 * q0 + c1 * q1 + c2 * q2;

    const bool  msk = best >= 0;
    const float mf  = msk ? 1.0f : 0.0f;
    const float gx  = msk ? (uu * 2.0f - 1.0f) : 0.0f;
    const float gy  = msk ? (qq * 2.0f - 1.0f) : 0.0f;

    // bilinear border sample from (B,3,TEXR,TEXR)
    const float ix = (gx + 1.0f) * (TEXR_ * 0.5f) - 0.5f;
    const float iy = (gy + 1.0f) * (TEXR_ * 0.5f) - 0.5f;
    const float ix0f = floorf(ix), iy0f = floorf(iy);
    const float fx = ix - ix0f,   fy = iy - iy0f;
    const int ix0 = (int)fminf(fmaxf(ix0f,        0.0f), (float)(TEXR_ - 1));
    const int ix1 = (int)fminf(fmaxf(ix0f + 1.0f, 0.0f), (float)(TEXR_ - 1));
    const int iy0 = (int)fminf(fmaxf(iy0f,        0.0f), (float)(TEXR_ - 1));
    const int iy1 = (int)fminf(fmaxf(iy0f + 1.0f, 0.0f), (float)(TEXR_ - 1));

    const size_t pixo = (size_t)b * PP + pix;
#pragma unroll
    for (int ch = 0; ch < 3; ++ch) {
        const float* tp = tex + (size_t)(b * 3 + ch) * (size_t)TEXR_ * TEXR_;
        const float v00 = tp[(size_t)iy0 * TEXR_ + ix0];
        const float v01 = tp[(size_t)iy0 * TEXR_ + ix1];
        const float v10 = tp[(size_t)iy1 * TEXR_ + ix0];
        const float v11 = tp[(size_t)iy1 * TEXR_ + ix1];
        const float s = v00 * (1.0f - fx) * (1.0f - fy) + v01 * fx * (1.0f - fy)
                      + v10 * (1.0f - fx) * fy          + v11 * fx * fy;
        out[OFF_RENDER + (size_t)(b * 3 + ch) * PP + pix] = s * mf;
    }
    out[OFF_MASK  + pixo] = mf;
    out[OFF_DEPTH + pixo] = msk ? (1.0f / invz) : 0.0f;
    out[OFF_IBUF  + pixo] = (float)best;
    out[OFF_VT + pixo * 2 + 0] = gx;
    out[OFF_VT + pixo * 2 + 1] = gy;
    out[OFF_BARY + (size_t)(b * 3 + 0) * PP + pix] = msk ? b0 : 0.0f;
    out[OFF_BARY + (size_t)(b * 3 + 1) * PP + pix] = msk ? b1 : 0.0f;
    out[OFF_BARY + (size_t)(b * 3 + 2) * PP + pix] = msk ? b2 : 0.0f;
}

// ---------------------------------------------------------------------------
extern "C" void kernel_launch(void* const* d_in, const int* in_sizes, int n_in,
                              void* d_out, int out_size, void* d_ws, size_t ws_size,
                              hipStream_t stream)
{
    const float* v       = (const float*)d_in[0];
    const float* tex     = (const float*)d_in[1];
    const float* vt      = (const float*)d_in[2];
    const float* campos  = (const float*)d_in[3];
    const float* camrot  = (const float*)d_in[4];
    const float* focal   = (const float*)d_in[5];
    const float* princpt = (const float*)d_in[6];
    const int*   vi      = (const int*)d_in[7];
    const int*   vti     = (const int*)d_in[8];
    float* out  = (float*)d_out;
    float* vpix = (float*)d_ws;               // B*V*3 floats = 96 KB scratch

    proj_wmma_kernel<<<(BB_ * VV_) / 16, 32, 0, stream>>>(
        v, campos, camrot, focal, princpt, vpix);

    raster_shade_kernel<<<BB_ * (PP / 256), 256, FF_ * 12 * sizeof(float), stream>>>(
        vpix, vi, vti, vt, tex, out);
}